// AttentiveGRU1_11287174053941
// MI455X (gfx1250) — compile-verified
//
#include <hip/hip_runtime.h>
#include <math.h>

typedef float v2f __attribute__((ext_vector_type(2)));
typedef float v4f __attribute__((ext_vector_type(4)));
typedef float v8f __attribute__((ext_vector_type(8)));

#define D 64
#define PAD 68  // row stride in LDS floats: conflict-free frag reads, 16B-aligned staging

__device__ __forceinline__ float sigmoidf_(float x) { return 1.0f / (1.0f + expf(-x)); }
__device__ __forceinline__ float eluf_(float x) { return x > 0.0f ? x : (expf(x) - 1.0f); }

// ---------------------------------------------------------------- init
__global__ void ag1_init(float* m, float* s, float* c, int N) {
    long idx = (long)blockIdx.x * blockDim.x + threadIdx.x;
    if (idx < N) {
        ((unsigned int*)m)[idx] = 0xFF800000u;  // -inf
        s[idx] = 0.0f;
    }
    if (idx < (long)N * D) c[idx] = 0.0f;
}

// ---------------------------------------------------------------- segment max (float atomic-max via sign split)
__global__ void ag1_segmax(const float* __restrict__ logits, const int* __restrict__ dst,
                           float* __restrict__ m, int E) {
    int e = blockIdx.x * blockDim.x + threadIdx.x;
    if (e >= E) return;
    float v = logits[e];
    int d = dst[e];
    if (v >= 0.0f) atomicMax((int*)(m + d), __float_as_int(v));
    else           atomicMin((unsigned int*)(m + d), __float_as_uint(v));
}

// ---------------------------------------------------------------- exp + segment sum
__global__ void ag1_expsum(const float* __restrict__ logits, const int* __restrict__ dst,
                           const float* __restrict__ m, float* __restrict__ a,
                           float* __restrict__ s, int E) {
    int e = blockIdx.x * blockDim.x + threadIdx.x;
    if (e >= E) return;
    int d = dst[e];
    float v = expf(logits[e] - m[d]);
    a[e] = v;
    atomicAdd(s + d, v);
}

// ---------------------------------------------------------------- WMMA helper: 16x16 tile, K=64
__device__ __forceinline__ v8f wmma_k64(const v2f af[16], const float* brow, int khalf) {
    v8f acc = {};
#pragma unroll
    for (int k0 = 0; k0 < 16; ++k0) {
        v2f b = *(const v2f*)(brow + 4 * k0 + 2 * khalf);
        acc = __builtin_amdgcn_wmma_f32_16x16x4_f32(false, af[k0], false, b, (short)0, acc,
                                                    false, false);
    }
    return acc;
}

// ---------------------------------------------------------------- edge transform + alpha scale + scatter-add
// block = 256 thr = 8 waves = 128 edges; LDS-staged We + A-tiles; T = A @ We^T; c[dst] += alpha*(T+b_e)
__global__ __launch_bounds__(256) void ag1_edge_gemm(
        const float* __restrict__ feats, const float* __restrict__ We,
        const float* __restrict__ be, const float* __restrict__ a_ws,
        const float* __restrict__ s_ws, const int* __restrict__ dst,
        float* __restrict__ c_ws, int E) {
    __shared__ float sWe[64 * PAD];
    __shared__ float sA[128 * PAD];
    __shared__ float sBe[64];
    __shared__ float sAlpha[128];
    __shared__ int   sDst[128];

    const int tid = threadIdx.x;
    const long blockE = (long)blockIdx.x * 128;

    // stage We (16KB) coalesced as b128
    for (int i = tid; i < 64 * 16; i += 256) {
        const int row = i >> 4, c4 = (i & 15) << 2;
        *(v4f*)(&sWe[row * PAD + c4]) = *(const v4f*)(We + row * D + c4);
    }
    if (tid < 64) sBe[tid] = be[tid];
    // stage 128-edge A tile (32KB) coalesced as b128
    for (int i = tid; i < 128 * 16; i += 256) {
        const int row = i >> 4, c4 = (i & 15) << 2;
        const long e = blockE + row;
        v4f v = {};
        if (e < E) v = *(const v4f*)(feats + e * D + c4);
        *(v4f*)(&sA[row * PAD + c4]) = v;
    }
    // per-edge alpha & destination, computed once
    if (tid < 128) {
        const long e = blockE + tid;
        if (e < E) {
            const int d = dst[e];
            sDst[tid] = d;
            sAlpha[tid] = a_ws[e] / s_ws[d];
        } else { sDst[tid] = 0; sAlpha[tid] = 0.0f; }
    }
    // prefetch next block's edge tile into L2 (global_prefetch_b8)
    if (blockE + 128 < E) __builtin_prefetch(feats + (blockE + 128) * D + tid * 4, 0, 1);
    __syncthreads();

    const int wave = tid >> 5, lane = tid & 31;
    const long ebase = blockE + (long)wave * 16;
    if (ebase >= E) return;                 // wave-uniform: EXEC all-ones at WMMA

    const int mrow = lane & 15, khalf = lane >> 4;
    const float* arow = &sA[(wave * 16 + mrow) * PAD];
    v2f af[16];
#pragma unroll
    for (int k0 = 0; k0 < 16; ++k0)
        af[k0] = *(const v2f*)(arow + 4 * k0 + 2 * khalf);

#pragma unroll
    for (int nb = 0; nb < 4; ++nb) {
        const int ncol = nb * 16 + (lane & 15);
        const float* brow = &sWe[ncol * PAD];   // B[k][n] = We[n][k]
        v8f acc = {};
#pragma unroll
        for (int k0 = 0; k0 < 16; ++k0) {
            v2f b = *(const v2f*)(brow + 4 * k0 + 2 * khalf);
            acc = __builtin_amdgcn_wmma_f32_16x16x4_f32(false, af[k0], false, b, (short)0, acc,
                                                        false, false);
        }
        const float bias = sBe[ncol];
#pragma unroll
        for (int v = 0; v < 8; ++v) {
            const int row = v + 8 * khalf;      // C/D layout: M = vgpr + 8*(lane>=16)
            const long e = ebase + row;
            if (e < E)
                atomicAdd(c_ws + (long)sDst[wave * 16 + row] * D + ncol,
                          sAlpha[wave * 16 + row] * (acc[v] + bias));
        }
    }
}

// ---------------------------------------------------------------- GRU cell: block = 128 thr = 4 waves = 64 nodes
__global__ __launch_bounds__(128) void ag1_node_gru(
        const float* __restrict__ c_ws, const float* __restrict__ h_in,
        const float* __restrict__ Wih, const float* __restrict__ Whh,
        const float* __restrict__ bih, const float* __restrict__ bhh,
        float* __restrict__ out, int N) {
    __shared__ float sC[64 * PAD];   // elu(context), 4 tiles of 16 rows
    __shared__ float sH[64 * PAD];   // node_feats

    const int tid = threadIdx.x;
    const long blockN = (long)blockIdx.x * 64;

    for (int i = tid; i < 64 * 16; i += 128) {
        const int row = i >> 4, c4 = (i & 15) << 2;
        const long n = blockN + row;
        v4f cv = {}, hv = {};
        if (n < N) {
            cv = *(const v4f*)(c_ws + n * D + c4);
            hv = *(const v4f*)(h_in + n * D + c4);
        }
        cv.x = eluf_(cv.x); cv.y = eluf_(cv.y); cv.z = eluf_(cv.z); cv.w = eluf_(cv.w);
        *(v4f*)(&sC[row * PAD + c4]) = cv;
        *(v4f*)(&sH[row * PAD + c4]) = hv;
    }
    __syncthreads();

    const int wave = tid >> 5, lane = tid & 31;
    const long nbase = blockN + (long)wave * 16;
    if (nbase >= N) return;                 // wave-uniform

    const int mrow = lane & 15, khalf = lane >> 4;
    const float* crow = &sC[(wave * 16 + mrow) * PAD];
    const float* hrow = &sH[(wave * 16 + mrow) * PAD];
    v2f ctxf[16], hf[16];
#pragma unroll
    for (int k0 = 0; k0 < 16; ++k0) {
        ctxf[k0] = *(const v2f*)(crow + 4 * k0 + 2 * khalf);
        hf[k0]   = *(const v2f*)(hrow + 4 * k0 + 2 * khalf);
    }

    for (int nb = 0; nb < 4; ++nb) {
        const int cg = nb * 16 + (lane & 15);          // column within one gate block
        // gi[:,j] = ctx @ Wih[j,:], j = gate*64+cg ; weights stream from L2 (96KB, hot)
        v8f i_r = wmma_k64(ctxf, Wih + (size_t)(0 * D + cg) * D, khalf);
        v8f h_r = wmma_k64(hf,   Whh + (size_t)(0 * D + cg) * D, khalf);
        v8f i_z = wmma_k64(ctxf, Wih + (size_t)(1 * D + cg) * D, khalf);
        v8f h_z = wmma_k64(hf,   Whh + (size_t)(1 * D + cg) * D, khalf);
        v8f i_n = wmma_k64(ctxf, Wih + (size_t)(2 * D + cg) * D, khalf);
        v8f h_n = wmma_k64(hf,   Whh + (size_t)(2 * D + cg) * D, khalf);

        const float br  = bih[0 * D + cg] + bhh[0 * D + cg];
        const float bz  = bih[1 * D + cg] + bhh[1 * D + cg];
        const float bni = bih[2 * D + cg];
        const float bnh = bhh[2 * D + cg];

#pragma unroll
        for (int v = 0; v < 8; ++v) {
            const int row = v + 8 * khalf;
            const long node = nbase + row;
            if (node < N) {
                const float r = sigmoidf_(i_r[v] + h_r[v] + br);
                const float z = sigmoidf_(i_z[v] + h_z[v] + bz);
                const float nn = tanhf(i_n[v] + bni + r * (h_n[v] + bnh));
                const float h = sH[(wave * 16 + row) * PAD + cg];
                const float hnew = (1.0f - z) * nn + z * h;
                out[node * D + cg] = fmaxf(hnew, 0.0f);
            }
        }
    }
}

// ---------------------------------------------------------------- launch
extern "C" void kernel_launch(void* const* d_in, const int* in_sizes, int n_in,
                              void* d_out, int out_size, void* d_ws, size_t ws_size,
                              hipStream_t stream) {
    const float* edge_logits = (const float*)d_in[0];
    const float* edge_feats  = (const float*)d_in[1];
    const float* node_feats  = (const float*)d_in[2];
    const int*   dst         = (const int*)d_in[3];
    const float* We  = (const float*)d_in[4];
    const float* be  = (const float*)d_in[5];
    const float* Wih = (const float*)d_in[6];
    const float* Whh = (const float*)d_in[7];
    const float* bih = (const float*)d_in[8];
    const float* bhh = (const float*)d_in[9];

    const int E = in_sizes[0];
    const int N = in_sizes[2] / D;

    float* ws = (float*)d_ws;
    float* m = ws;               // [N]
    float* s = ws + N;           // [N]
    float* a = ws + 2 * (long)N; // [E]
    float* c = a + E;            // [N*D]

    const long initTot = (long)N * D;
    ag1_init<<<(int)((initTot + 255) / 256), 256, 0, stream>>>(m, s, c, N);
    ag1_segmax<<<(E + 255) / 256, 256, 0, stream>>>(edge_logits, dst, m, E);
    ag1_expsum<<<(E + 255) / 256, 256, 0, stream>>>(edge_logits, dst, m, a, s, E);

    ag1_edge_gemm<<<(E + 127) / 128, 256, 0, stream>>>(edge_feats, We, be, a, s, dst, c, E);

    ag1_node_gru<<<(N + 63) / 64, 128, 0, stream>>>(c, node_feats, Wih, Whh, bih, bhh,
                                                    (float*)d_out, N);
}